// QuantumNATHybridModel_65481071397141
// MI455X (gfx1250) — compile-verified
//
#include <hip/hip_runtime.h>
#include <hip/hip_bf16.h>

// ---------------- CDNA5 WMMA types ----------------
typedef __attribute__((ext_vector_type(2))) float v2f;
typedef __attribute__((ext_vector_type(8))) float v8f;

// async-copy payload type per hipcc diagnostic: v4i (16B) pointers
typedef int v4i_ __attribute__((vector_size(16)));
typedef __attribute__((address_space(1))) v4i_* gv4i_ptr;
typedef __attribute__((address_space(3))) v4i_* lv4i_ptr;

#define N_QUBITS 4
#define N_LAYERS 3
#define BATCH 65536
#define NPIX 576              // 24*24
#define ROWS_PER_BLOCK 256
#define CHUNK 16              // samples staged per LDS pass
#define POOL_STRIDE 17        // padded stride for s_pooled (bank-conflict avoidance)
#define BN_EPS 1e-5f

#if __has_builtin(__builtin_amdgcn_global_load_async_to_lds_b128)
#define USE_ASYNC_LDS 1
#else
#define USE_ASYNC_LDS 0
#endif

__device__ __forceinline__ void wait_asynccnt0() {
#if __has_builtin(__builtin_amdgcn_s_wait_asynccnt)
    __builtin_amdgcn_s_wait_asynccnt(0);
#else
    asm volatile("s_wait_asynccnt 0" ::: "memory");
#endif
}

// ============================================================================
// Kernel 1: pooling + WMMA encode + 4-qubit circuit + BN partial sums
// ============================================================================
__global__ __launch_bounds__(256) void qnat_main_kernel(
    const float* __restrict__ x,        // [B, 576]
    const float* __restrict__ enc_w,    // [4, 16]
    const float* __restrict__ enc_b,    // [4]
    const float* __restrict__ qparams,  // [3, 4, 3]
    float* __restrict__ out,            // [B, 4]  (pre-BN after this kernel)
    float* __restrict__ partials)       // [gridDim.x, 8] (sum z, sum z^2 per qubit)
{
    __shared__ float4 s_stage4[CHUNK * NPIX / 4];            // 36864 B
    __shared__ float  s_pooled[ROWS_PER_BLOCK * POOL_STRIDE];// 17408 B
    __shared__ float  s_ang[ROWS_PER_BLOCK * 4];             //  4096 B
    __shared__ float  s_encw[256];                           //  1024 B (zero-padded 16x16)
    __shared__ float  s_qtrig[N_LAYERS * N_QUBITS * 4];      //   192 B {cry,sry,crz,srz}
    __shared__ float  s_red[8 * 8];                          //   256 B

    const int tid  = threadIdx.x;
    const int wave = tid >> 5;
    const int lane = tid & 31;
    const int row0 = blockIdx.x * ROWS_PER_BLOCK;
    float* s_stage = (float*)s_stage4;

    // Stage enc_w as a zero-padded 16x16 image: s_encw[n*16 + k] = (n<4)?enc_w[n][k]:0.
    // This keeps the WMMA B-operand feed branch-free (EXEC stays all-ones).
    {
        const int n = tid >> 4, k = tid & 15;
        s_encw[tid] = (n < N_QUBITS) ? enc_w[n * 16 + k] : 0.f;
    }
    if (tid < N_LAYERS * N_QUBITS) {
        const int l = tid >> 2, q = tid & 3;
        const float p0 = qparams[(l * 4 + q) * 3 + 0];
        const float p1 = qparams[(l * 4 + q) * 3 + 1];
        s_qtrig[tid * 4 + 0] = cosf(0.5f * p0);
        s_qtrig[tid * 4 + 1] = sinf(0.5f * p0);
        s_qtrig[tid * 4 + 2] = cosf(0.5f * p1);
        s_qtrig[tid * 4 + 3] = sinf(0.5f * p1);
    }

    // ---------------- Phase A: staged, coalesced pooling ----------------
    for (int ch = 0; ch < ROWS_PER_BLOCK / CHUNK; ++ch) {
        __syncthreads();  // previous chunk fully consumed
        const float4* src = (const float4*)(x + (size_t)(row0 + ch * CHUNK) * NPIX);
#if USE_ASYNC_LDS
        // CDNA5 async copy: global -> LDS without a VGPR round trip (ASYNCcnt).
        #pragma unroll
        for (int i = 0; i < (CHUNK * NPIX / 4) / 256; ++i) {
            __builtin_amdgcn_global_load_async_to_lds_b128(
                (gv4i_ptr)(src + tid + i * 256),
                (lv4i_ptr)&s_stage4[tid + i * 256],
                0, 0);
        }
        wait_asynccnt0();
#else
        #pragma unroll
        for (int i = 0; i < (CHUNK * NPIX / 4) / 256; ++i)
            s_stage4[tid + i * 256] = src[tid + i * 256];
#endif
        __syncthreads();

        // one pooled value per thread: 16 samples x 16 pool windows
        const int ls = tid >> 4;          // sample within chunk
        const int p  = tid & 15;          // pooled index
        const int py = p >> 2, px = p & 3;
        const float* base = s_stage + ls * NPIX + py * 6 * 24 + px * 6;
        float acc = 0.f;
        #pragma unroll
        for (int r = 0; r < 6; ++r)
            #pragma unroll
            for (int c = 0; c < 6; ++c)
                acc += base[r * 24 + c];
        s_pooled[(ch * CHUNK + ls) * POOL_STRIDE + p] = acc * (1.0f / 36.0f);
    }
    __syncthreads();

    // ---------------- Phase B: encode via V_WMMA_F32_16X16X4_F32 ----------------
    // angles[m, n] = sum_k pooled[m, k] * enc_w[n, k];  K=16 -> 4 chained WMMAs.
    // A 16x4 f32 layout: lanes 0-15 hold {K=0,1}, lanes 16-31 hold {K=2,3} (ISA 7.12.2).
    // B 4x16 f32: VGPR0 = rows K=0 (lanes 0-15) / K=2 (lanes 16-31); VGPR1 = K=1 / K=3.
    const int mrow  = lane & 15;
    const int kHalf = (lane < 16) ? 0 : 2;
    #pragma unroll
    for (int t = 0; t < 2; ++t) {
        const int m0 = wave * 32 + t * 16;
        v8f acc = {0.f, 0.f, 0.f, 0.f, 0.f, 0.f, 0.f, 0.f};
        #pragma unroll
        for (int kc = 0; kc < 4; ++kc) {
            const float* prow = &s_pooled[(m0 + mrow) * POOL_STRIDE + kc * 4 + kHalf];
            v2f a; a.x = prow[0]; a.y = prow[1];
            v2f b; b.x = s_encw[mrow * 16 + kc * 4 + kHalf + 0];
                   b.y = s_encw[mrow * 16 + kc * 4 + kHalf + 1];
            acc = __builtin_amdgcn_wmma_f32_16x16x4_f32(
                false, a, false, b, (short)0, acc, false, false);
        }
        // D layout: VGPR r -> M=r (lanes 0-15, N=lane) and M=r+8 (lanes 16-31, N=lane-16)
        if (lane < 4) {
            #pragma unroll
            for (int r = 0; r < 8; ++r) s_ang[(m0 + r) * 4 + lane] = acc[r];
        } else if (lane >= 16 && lane < 20) {
            #pragma unroll
            for (int r = 0; r < 8; ++r) s_ang[(m0 + 8 + r) * 4 + (lane - 16)] = acc[r];
        }
    }
    __syncthreads();

    // ---------------- Phase C: per-thread 4-qubit statevector ----------------
    float ang[4];
    #pragma unroll
    for (int q = 0; q < 4; ++q) ang[q] = s_ang[tid * 4 + q] + enc_b[q];

    // Initial state = RX(ang0..3) |0000>: product state, amp[i] = prod * (-i)^popcount(i)
    float cq[4], sq[4];
    #pragma unroll
    for (int q = 0; q < 4; ++q) { cq[q] = cosf(0.5f * ang[q]); sq[q] = sinf(0.5f * ang[q]); }

    float re[16], im[16];
    #pragma unroll
    for (int i = 0; i < 16; ++i) {
        const float m = ((i & 8) ? sq[0] : cq[0]) * ((i & 4) ? sq[1] : cq[1]) *
                        ((i & 2) ? sq[2] : cq[2]) * ((i & 1) ? sq[3] : cq[3]);
        const int k = __builtin_popcount(i) & 3;   // compile-time after unroll
        re[i] = (k == 0) ? m : ((k == 2) ? -m : 0.f);
        im[i] = (k == 1) ? -m : ((k == 3) ? m : 0.f);
    }

    #pragma unroll
    for (int l = 0; l < N_LAYERS; ++l) {
        #pragma unroll
        for (int q = 0; q < 4; ++q) {
            const float cry = s_qtrig[(l * 4 + q) * 4 + 0];
            const float sry = s_qtrig[(l * 4 + q) * 4 + 1];
            const float crz = s_qtrig[(l * 4 + q) * 4 + 2];
            const float srz = s_qtrig[(l * 4 + q) * 4 + 3];
            const int st = 8 >> q;
            #pragma unroll
            for (int i = 0; i < 16; ++i) {
                if (i & st) continue;
                const int j = i | st;
                const float r0 = re[i], i0 = im[i], r1 = re[j], i1 = im[j];
                // RY
                const float nr0 = cry * r0 - sry * r1, ni0 = cry * i0 - sry * i1;
                const float nr1 = sry * r0 + cry * r1, ni1 = sry * i0 + cry * i1;
                // RZ: |0> *= e^{-i t/2}, |1> *= e^{+i t/2}
                re[i] = crz * nr0 + srz * ni0;  im[i] = crz * ni0 - srz * nr0;
                re[j] = crz * nr1 - srz * ni1;  im[j] = crz * ni1 + srz * nr1;
            }
        }
        // CNOT ring: (0,1),(1,2),(2,3),(3,0)
        #pragma unroll
        for (int e = 0; e < 4; ++e) {
            const int mc = 8 >> e, mt = 8 >> ((e + 1) & 3);
            #pragma unroll
            for (int i = 0; i < 16; ++i) {
                if ((i & mc) && !(i & mt)) {
                    const int j = i | mt;
                    float tr = re[i]; re[i] = re[j]; re[j] = tr;
                    float ti = im[i]; im[i] = im[j]; im[j] = ti;
                }
            }
        }
    }

    // Z expectations
    float z[4];
    #pragma unroll
    for (int q = 0; q < 4; ++q) {
        const int m = 8 >> q;
        float s = 0.f;
        #pragma unroll
        for (int i = 0; i < 16; ++i) {
            const float p = re[i] * re[i] + im[i] * im[i];
            s += (i & m) ? -p : p;
        }
        z[q] = s;
    }
    *(float4*)(out + (size_t)(row0 + tid) * 4) = make_float4(z[0], z[1], z[2], z[3]);

    // ---------------- Phase D: deterministic BN partial sums ----------------
    float v[8] = {z[0], z[1], z[2], z[3],
                  z[0] * z[0], z[1] * z[1], z[2] * z[2], z[3] * z[3]};
    #pragma unroll
    for (int k = 0; k < 8; ++k) {
        #pragma unroll
        for (int off = 16; off >= 1; off >>= 1)
            v[k] += __shfl_xor(v[k], off, 32);
    }
    if (lane == 0) {
        #pragma unroll
        for (int k = 0; k < 8; ++k) s_red[wave * 8 + k] = v[k];
    }
    __syncthreads();
    if (tid < 8) {
        float s = 0.f;
        #pragma unroll
        for (int w = 0; w < 8; ++w) s += s_red[w * 8 + tid];
        partials[blockIdx.x * 8 + tid] = s;
    }
}

// ============================================================================
// Kernel 2: fold per-block partials -> BN scale/bias
// ============================================================================
__global__ __launch_bounds__(64) void qnat_bn_reduce_kernel(
    const float* __restrict__ partials,  // [256, 8]
    const float* __restrict__ gamma,
    const float* __restrict__ beta,
    float* __restrict__ coef)            // [8]: scale[4], bias[4]
{
    __shared__ float sums[8];
    const int t = threadIdx.x;
    if (t < 8) {
        float s = 0.f;
        for (int i = 0; i < 256; ++i) s += partials[i * 8 + t];
        sums[t] = s;
    }
    __syncthreads();
    if (t < 4) {
        const float inv_b = 1.0f / (float)BATCH;
        const float mean  = sums[t] * inv_b;
        const float var   = sums[4 + t] * inv_b - mean * mean;
        const float sc    = gamma[t] * rsqrtf(var + BN_EPS);
        coef[t]     = sc;
        coef[4 + t] = beta[t] - mean * sc;
    }
}

// ============================================================================
// Kernel 3: apply BN in place on d_out
// ============================================================================
__global__ __launch_bounds__(256) void qnat_bn_apply_kernel(
    float* __restrict__ out, const float* __restrict__ coef)
{
    const int i = blockIdx.x * 256 + threadIdx.x;
    const int q = i & 3;
    out[i] = out[i] * coef[q] + coef[4 + q];
}

// ============================================================================
extern "C" void kernel_launch(void* const* d_in, const int* in_sizes, int n_in,
                              void* d_out, int out_size, void* d_ws, size_t ws_size,
                              hipStream_t stream) {
    const float* x       = (const float*)d_in[0];
    const float* enc_w   = (const float*)d_in[1];
    const float* enc_b   = (const float*)d_in[2];
    const float* qparams = (const float*)d_in[3];
    const float* gamma   = (const float*)d_in[4];
    const float* beta    = (const float*)d_in[5];
    float* outp = (float*)d_out;

    float* partials = (float*)d_ws;              // 256 * 8 floats
    float* coef     = partials + 256 * 8;        // 8 floats

    const int nblocks = BATCH / ROWS_PER_BLOCK;  // 256
    qnat_main_kernel<<<nblocks, 256, 0, stream>>>(x, enc_w, enc_b, qparams, outp, partials);
    qnat_bn_reduce_kernel<<<1, 64, 0, stream>>>(partials, gamma, beta, coef);
    qnat_bn_apply_kernel<<<(BATCH * N_QUBITS) / 256, 256, 0, stream>>>(outp, coef);
}